// ComplementConstraint_7825430413733
// MI455X (gfx1250) — compile-verified
//
#include <hip/hip_runtime.h>

// Leave-one-out logsumexp, B x K (K = 1000), f32 in / f32 out.
//   m = max_k x[b,k];  s = sum_k exp(x[b,k]-m);  out[b,k] = -(m + log(s - exp(x[b,k]-m)))
//
// MI455X reasoning: pure streaming op, 1.05 GB moved per call -> ~45us HBM floor
// at 23.3 TB/s. Wave32-per-row keeps the entire row in VGPRs (no LDS, no
// barriers), reductions are 5-step shfl_xor butterflies, loads/stores are
// 128-bit non-temporal (working set >> 192MB L2).

typedef float v4f __attribute__((ext_vector_type(4)));

#define KCLS        1000
#define V4_PER_ROW  250          // 1000 / 4
#define FULL_CHUNKS 7            // chunks 0..6: 7*32 = 224 float4s
#define TAIL_V4     26           // chunk 7: lanes 0..25 -> 224+26 = 250 float4s
#define WAVES_PER_BLOCK 8

__device__ __forceinline__ float wave_max32(float v) {
#pragma unroll
  for (int off = 16; off > 0; off >>= 1)
    v = fmaxf(v, __shfl_xor(v, off, 32));
  return v;
}

__device__ __forceinline__ float wave_sum32(float v) {
#pragma unroll
  for (int off = 16; off > 0; off >>= 1)
    v += __shfl_xor(v, off, 32);
  return v;
}

__global__ __launch_bounds__(32 * WAVES_PER_BLOCK)
void loo_lse_kernel(const float* __restrict__ logits,
                    float* __restrict__ out,
                    int B) {
  const int wave = threadIdx.x >> 5;
  const int lane = threadIdx.x & 31;
  const int row  = blockIdx.x * WAVES_PER_BLOCK + wave;
  if (row >= B) return;

  const v4f* __restrict__ src = (const v4f*)(logits + (size_t)row * KCLS);
  v4f*       __restrict__ dst = (v4f*)(out + (size_t)row * KCLS);

  const float L2E = 1.4426950408889634f;   // log2(e)
  const float LN2 = 0.6931471805599453f;   // ln(2)

  // ---- Load entire row into registers (coalesced b128, non-temporal) ----
  v4f x[FULL_CHUNKS + 1];
#pragma unroll
  for (int i = 0; i < FULL_CHUNKS; ++i)
    x[i] = __builtin_nontemporal_load(&src[i * 32 + lane]);
  const bool has_tail = (lane < TAIL_V4);
  if (has_tail)
    x[FULL_CHUNKS] = __builtin_nontemporal_load(&src[FULL_CHUNKS * 32 + lane]);

  // ---- Pass 1: row max ----
  float m = -__builtin_inff();
#pragma unroll
  for (int i = 0; i < FULL_CHUNKS; ++i) {
#pragma unroll
    for (int j = 0; j < 4; ++j)
      m = fmaxf(m, x[i][j]);
  }
  if (has_tail) {
#pragma unroll
    for (int j = 0; j < 4; ++j)
      m = fmaxf(m, x[FULL_CHUNKS][j]);
  }
  m = wave_max32(m);

  // ---- Pass 2: sum of exp(x - m) ----
  float s = 0.0f;
#pragma unroll
  for (int i = 0; i < FULL_CHUNKS; ++i) {
#pragma unroll
    for (int j = 0; j < 4; ++j)
      s += __builtin_amdgcn_exp2f((x[i][j] - m) * L2E);
  }
  if (has_tail) {
#pragma unroll
    for (int j = 0; j < 4; ++j)
      s += __builtin_amdgcn_exp2f((x[FULL_CHUNKS][j] - m) * L2E);
  }
  s = wave_sum32(s);

  // ---- Pass 3: out = -(m + log(s - e_k)), streamed back non-temporal ----
#pragma unroll
  for (int i = 0; i < FULL_CHUNKS; ++i) {
    v4f r;
#pragma unroll
    for (int j = 0; j < 4; ++j) {
      float e = __builtin_amdgcn_exp2f((x[i][j] - m) * L2E);
      r[j] = -(m + LN2 * __builtin_amdgcn_logf(s - e));
    }
    __builtin_nontemporal_store(r, &dst[i * 32 + lane]);
  }
  if (has_tail) {
    v4f r;
#pragma unroll
    for (int j = 0; j < 4; ++j) {
      float e = __builtin_amdgcn_exp2f((x[FULL_CHUNKS][j] - m) * L2E);
      r[j] = -(m + LN2 * __builtin_amdgcn_logf(s - e));
    }
    __builtin_nontemporal_store(r, &dst[FULL_CHUNKS * 32 + lane]);
  }
}

extern "C" void kernel_launch(void* const* d_in, const int* in_sizes, int n_in,
                              void* d_out, int out_size, void* d_ws, size_t ws_size,
                              hipStream_t stream) {
  (void)n_in; (void)out_size; (void)d_ws; (void)ws_size;
  const float* logits = (const float*)d_in[0];
  float* out = (float*)d_out;

  const int B = in_sizes[0] / KCLS;   // 131072 rows
  const int blocks = (B + WAVES_PER_BLOCK - 1) / WAVES_PER_BLOCK;

  loo_lse_kernel<<<blocks, 32 * WAVES_PER_BLOCK, 0, stream>>>(logits, out, B);
}